// ELGCA_283467841771
// MI455X (gfx1250) — compile-verified
//
#include <hip/hip_runtime.h>
#include <hip/hip_bf16.h>
#include <math.h>

// ELGCA fused pipeline for MI455X (gfx1250, wave32).
// fp32 end-to-end; matrix work on V_WMMA_F32_16X16X4_F32 (HBM-bound workload,
// ~11 FLOP/byte, so full-precision matrix path is the right choice).

typedef __attribute__((ext_vector_type(2))) float v2f;
typedef __attribute__((ext_vector_type(8))) float v8f;

#define B_   16
#define C_   128
#define H_   128
#define W_   128
#define HW_  16384
#define C2_  64      // x2 input channels
#define OC_  128     // 1x1 conv output channels
#define PQ_  4096    // pooled pixels (64*64)
#define WSTR 68      // LDS row stride (floats) for weight tile: avoids 64-bank
                     // conflicts on per-lane ds_load_b64 A-fragment reads

__device__ __forceinline__ float gelu_f(float v) {
  return 0.5f * v * (1.0f + erff(v * 0.7071067811865475f));
}

__device__ __forceinline__ v8f vz8() {
  v8f z = {0.f, 0.f, 0.f, 0.f, 0.f, 0.f, 0.f, 0.f};
  return z;
}

// D = A(16x4 f32) * B(4x16 f32) + C(16x16 f32)
__device__ __forceinline__ v8f wmma4(v2f a, v2f b, v8f c) {
  return __builtin_amdgcn_wmma_f32_16x16x4_f32(false, a, false, b, (short)0, c,
                                               false, false);
}

// ---------------------------------------------------------------------------
// Kernel 1: depthwise 3x3 + GELU on channels [0,64) -> out channels [0,64)
// 16x16 output tile per block, 18x19 LDS halo tile (padded row to dodge banks)
// ---------------------------------------------------------------------------
__global__ __launch_bounds__(256) void k_dwconv_gelu(
    const float* __restrict__ x, const float* __restrict__ w,
    const float* __restrict__ bias, float* __restrict__ out)
{
  __shared__ float tile[18 * 19];
  const int tid = threadIdx.x;
  const int tx = tid & 15, ty = tid >> 4;
  const int plane = blockIdx.z;               // b*64 + ch
  const int b = plane >> 6, ch = plane & 63;
  const int x0 = blockIdx.x << 4, y0 = blockIdx.y << 4;
  const float* src = x + ((size_t)b * C_ + ch) * HW_;

  for (int i = tid; i < 18 * 18; i += 256) {
    const int ly = i / 18, lx = i - ly * 18;
    const int gy = y0 + ly - 1, gx = x0 + lx - 1;
    float v = 0.f;
    if ((unsigned)gy < (unsigned)H_ && (unsigned)gx < (unsigned)W_)
      v = src[gy * W_ + gx];
    tile[ly * 19 + lx] = v;
  }
  __syncthreads();

  const float* wc = w + ch * 9;
  float s = bias[ch];
#pragma unroll
  for (int dy = 0; dy < 3; ++dy)
#pragma unroll
    for (int dx = 0; dx < 3; ++dx)
      s = fmaf(wc[dy * 3 + dx], tile[(ty + dy) * 19 + tx + dx], s);

  out[((size_t)b * C_ + ch) * HW_ + (size_t)(y0 + ty) * W_ + (x0 + tx)] =
      gelu_f(s);
}

// ---------------------------------------------------------------------------
// Kernel 2: 1x1 conv (GEMM 128x64 * 64xN) via WMMA f32 16x16x4 + GELU.
// Per wave: 16 pixels x 128 out-channels = 8 M-tiles x 16 K-steps.
// Weights staged in LDS (row stride WSTR=68 floats). Epilogue routes heads:
//   head0->q_ws, head1->k_ws, head2->v_ws, head3->out channels [64,96).
// ---------------------------------------------------------------------------
__global__ __launch_bounds__(256) void k_conv1x1_wmma(
    const float* __restrict__ x, const float* __restrict__ w,
    const float* __restrict__ bias,
    float* __restrict__ qw, float* __restrict__ kw, float* __restrict__ vw,
    float* __restrict__ out)
{
  __shared__ float wl[OC_ * WSTR];
  __shared__ float bl[OC_];
  const int tid = threadIdx.x;

  for (int i = tid; i < OC_ * C2_; i += 256) {
    const int oc = i >> 6, ic = i & 63;
    wl[oc * WSTR + ic] = w[i];
  }
  if (tid < OC_) bl[tid] = bias[tid];

  const int b    = blockIdx.x >> 7;
  const int px0  = ((blockIdx.x & 127) << 7) + ((tid >> 5) << 4);
  const int lane = tid & 31, lr = lane & 15, hs = lane >> 4;

  const float* xb = x + ((size_t)b * C_ + C2_) * HW_;

  // B fragments: X[ic, px], K along ic.  Lane layout per ISA A/B tables.
  v2f bfr[16];
#pragma unroll
  for (int kk = 0; kk < 16; ++kk) {
    const int ic = (kk << 2) + (hs << 1);
    v2f t;
    t.x = xb[(size_t)ic * HW_ + px0 + lr];
    t.y = xb[(size_t)(ic + 1) * HW_ + px0 + lr];
    bfr[kk] = t;
  }
  __syncthreads();

  v8f acc[8];
#pragma unroll
  for (int t = 0; t < 8; ++t) acc[t] = vz8();

#pragma unroll
  for (int kk = 0; kk < 16; ++kk) {
#pragma unroll
    for (int t = 0; t < 8; ++t) {
      const v2f a =
          *(const v2f*)&wl[((t << 4) + lr) * WSTR + (kk << 2) + (hs << 1)];
      acc[t] = wmma4(a, bfr[kk], acc[t]);
    }
  }

  const size_t b32 = (size_t)b * 32 * HW_;
#pragma unroll
  for (int t = 0; t < 8; ++t) {
    const int head = t >> 1;  // compile-time after unroll
#pragma unroll
    for (int r = 0; r < 8; ++r) {
      const int oc  = (t << 4) + r + (hs << 3);
      const int c32 = oc & 31;
      const float g = gelu_f(acc[t][r] + bl[oc]);
      const size_t off = b32 + (size_t)c32 * HW_ + px0 + lr;
      if (head == 0)      qw[off] = g;
      else if (head == 1) kw[off] = g;
      else if (head == 2) vw[off] = g;
      else out[((size_t)b * C_ + 64 + c32) * HW_ + px0 + lr] = g;
    }
  }
}

// ---------------------------------------------------------------------------
// Kernel 3: qf = avgpool3x3 s2 p1 (zero-pad, always /9) ; kf = maxpool2x2 s2
// One thread per pooled pixel (computes both).
// ---------------------------------------------------------------------------
__global__ __launch_bounds__(256) void k_pool(
    const float* __restrict__ q, const float* __restrict__ k,
    float* __restrict__ qf, float* __restrict__ kf)
{
  const int idx = blockIdx.x * 256 + threadIdx.x;   // exactly 16*32*4096 threads
  const int p  = idx & (PQ_ - 1);
  const int bc = idx >> 12;
  const int ho = p >> 6, wo = p & 63;
  const float* qs = q + (size_t)bc * HW_;
  const float* ks = k + (size_t)bc * HW_;

  float s = 0.f;
#pragma unroll
  for (int dy = 0; dy < 3; ++dy) {
    const int hh = 2 * ho - 1 + dy;
#pragma unroll
    for (int dx = 0; dx < 3; ++dx) {
      const int ww = 2 * wo - 1 + dx;
      if ((unsigned)hh < 128u && (unsigned)ww < 128u) s += qs[hh * W_ + ww];
    }
  }
  qf[idx] = s * (1.f / 9.f);

  const float* mp = ks + (ho << 1) * W_ + (wo << 1);
  kf[idx] = fmaxf(fmaxf(mp[0], mp[1]), fmaxf(mp[W_], mp[W_ + 1]));
}

// ---------------------------------------------------------------------------
// Kernel 4: qk[c,d] = sum_n qf[c,n]*kf[d,n] (32x32, K=4096) via WMMA,
// 8 waves split K, LDS reduce, softmax over c, write attn transposed:
// attn[d*32+c] = exp(qk[c,d]-m_d)/sum_c.  One block per batch.
// ---------------------------------------------------------------------------
__global__ __launch_bounds__(256) void k_qk_softmax(
    const float* __restrict__ qf, const float* __restrict__ kf,
    float* __restrict__ attn)
{
  __shared__ float red[8 * 1024];
  const int b = blockIdx.x;
  const int tid = threadIdx.x;
  const int lane = tid & 31, wv = tid >> 5;
  const int lr = lane & 15, hs = lane >> 4;
  const int ko = hs << 1;
  const float* qp = qf + (size_t)b * 32 * PQ_;
  const float* kp = kf + (size_t)b * 32 * PQ_;

  v8f a00 = vz8(), a01 = vz8(), a10 = vz8(), a11 = vz8();
  const int n0 = wv << 9;
  for (int n = n0; n < n0 + 512; n += 4) {
    // contiguous 8B loads: K runs along contiguous n axis for both A and B
    const v2f qa0 = *(const v2f*)(qp + (size_t)lr * PQ_ + n + ko);
    const v2f qa1 = *(const v2f*)(qp + (size_t)(16 + lr) * PQ_ + n + ko);
    const v2f kb0 = *(const v2f*)(kp + (size_t)lr * PQ_ + n + ko);
    const v2f kb1 = *(const v2f*)(kp + (size_t)(16 + lr) * PQ_ + n + ko);
    a00 = wmma4(qa0, kb0, a00);
    a01 = wmma4(qa0, kb1, a01);
    a10 = wmma4(qa1, kb0, a10);
    a11 = wmma4(qa1, kb1, a11);
  }

#pragma unroll
  for (int r = 0; r < 8; ++r) {
    const int c = r + (hs << 3);
    red[wv * 1024 + c * 32 + lr]             = a00[r];
    red[wv * 1024 + c * 32 + 16 + lr]        = a01[r];
    red[wv * 1024 + (16 + c) * 32 + lr]      = a10[r];
    red[wv * 1024 + (16 + c) * 32 + 16 + lr] = a11[r];
  }
  __syncthreads();

  for (int i = tid; i < 1024; i += 256) {
    float s = 0.f;
#pragma unroll
    for (int wi = 0; wi < 8; ++wi) s += red[wi * 1024 + i];
    red[i] = s;
  }
  __syncthreads();

  if (tid < 32) {
    const int d = tid;
    float m = -3.4e38f;
#pragma unroll
    for (int c = 0; c < 32; ++c) m = fmaxf(m, red[c * 32 + d]);
    float e[32];
    float sum = 0.f;
#pragma unroll
    for (int c = 0; c < 32; ++c) { e[c] = __expf(red[c * 32 + d] - m); sum += e[c]; }
    const float inv = 1.f / sum;
#pragma unroll
    for (int c = 0; c < 32; ++c)
      attn[(size_t)b * 1024 + d * 32 + c] = e[c] * inv;
  }
}

// ---------------------------------------------------------------------------
// Kernel 5: out2 = attn(32x32) * v(32x16384) via WMMA -> out channels [96,128)
// attn fragments in registers; per wave: 16 px, 2 M-tiles x 8 K-steps.
// ---------------------------------------------------------------------------
__global__ __launch_bounds__(256) void k_out2_wmma(
    const float* __restrict__ attn, const float* __restrict__ v,
    float* __restrict__ out)
{
  const int tid = threadIdx.x;
  const int b   = blockIdx.x >> 7;
  const int px0 = ((blockIdx.x & 127) << 7) + ((tid >> 5) << 4);
  const int lane = tid & 31, lr = lane & 15, hs = lane >> 4;
  const float* A  = attn + (size_t)b * 1024;
  const float* vb = v + (size_t)b * 32 * HW_;

  v2f af[2][8];
#pragma unroll
  for (int mt = 0; mt < 2; ++mt)
#pragma unroll
    for (int kk = 0; kk < 8; ++kk)
      af[mt][kk] =
          *(const v2f*)(A + ((mt << 4) + lr) * 32 + (kk << 2) + (hs << 1));

  v8f acc0 = vz8(), acc1 = vz8();
#pragma unroll
  for (int kk = 0; kk < 8; ++kk) {
    const int c = (kk << 2) + (hs << 1);
    v2f bf;
    bf.x = vb[(size_t)c * HW_ + px0 + lr];
    bf.y = vb[(size_t)(c + 1) * HW_ + px0 + lr];
    acc0 = wmma4(af[0][kk], bf, acc0);
    acc1 = wmma4(af[1][kk], bf, acc1);
  }

  float* ob = out + ((size_t)b * C_ + 96) * HW_;
#pragma unroll
  for (int r = 0; r < 8; ++r) {
    const int m = r + (hs << 3);
    ob[(size_t)m * HW_ + px0 + lr]        = acc0[r];
    ob[(size_t)(16 + m) * HW_ + px0 + lr] = acc1[r];
  }
}

// ---------------------------------------------------------------------------
extern "C" void kernel_launch(void* const* d_in, const int* in_sizes, int n_in,
                              void* d_out, int out_size, void* d_ws,
                              size_t ws_size, hipStream_t stream)
{
  (void)in_sizes; (void)n_in; (void)out_size; (void)ws_size;
  const float* x      = (const float*)d_in[0];
  const float* dw_w   = (const float*)d_in[1];
  const float* dw_b   = (const float*)d_in[2];
  const float* qkvl_w = (const float*)d_in[3];
  const float* qkvl_b = (const float*)d_in[4];
  float* out = (float*)d_out;

  // Workspace layout (floats): q,k,v (16*32*16384 each), qf,kf (16*32*4096),
  // attn (16*1024).  Total ~118 MB.
  float* ws = (float*)d_ws;
  const size_t SZ = (size_t)B_ * 32 * HW_;
  const size_t PZ = (size_t)B_ * 32 * PQ_;
  float* q_ws    = ws;
  float* k_ws    = q_ws + SZ;
  float* v_ws    = k_ws + SZ;
  float* qf_ws   = v_ws + SZ;
  float* kf_ws   = qf_ws + PZ;
  float* attn_ws = kf_ws + PZ;

  k_dwconv_gelu<<<dim3(8, 8, B_ * 64), 256, 0, stream>>>(x, dw_w, dw_b, out);
  k_conv1x1_wmma<<<dim3(B_ * 128), 256, 0, stream>>>(x, qkvl_w, qkvl_b,
                                                     q_ws, k_ws, v_ws, out);
  k_pool<<<dim3((B_ * 32 * PQ_) / 256), 256, 0, stream>>>(q_ws, k_ws,
                                                          qf_ws, kf_ws);
  k_qk_softmax<<<dim3(B_), 256, 0, stream>>>(qf_ws, kf_ws, attn_ws);
  k_out2_wmma<<<dim3(B_ * 128), 256, 0, stream>>>(attn_ws, v_ws, out);
}